// MultiHeadAttentionQuantum_69552700392075
// MI455X (gfx1250) — compile-verified
//
#include <hip/hip_runtime.h>

// MultiHeadAttentionQuantum for MI455X (gfx1250, wave32).
//
// Quantum circuit reduced analytically (1 layer, RX + CNOT ring, <Z_j>):
//   c_i = cos(x_i + w_i);  z_0 = c1*...*c7;  z_j = c0*...*c_j (j>=1)
//
// Attention: H=2, d=4 -> exact fit for V_WMMA_F32_16X16X4_F32.
// Dominant cost is the 268 MB attn output write (HBM-bound, ~12us floor);
// logits are recomputed per pass (compute is ~0.5% of roofline).

#define NBATCH 8
#define NSEQ   2048
#define NEMB   8
#define NHEAD  2
#define NDIM   4
#define NKTILES (NSEQ / 16)

typedef __attribute__((ext_vector_type(2))) float v2f;
typedef __attribute__((ext_vector_type(8))) float v8f;

// ---------------------------------------------------------------------------
// Quantum "circuit" projection: one row (8 floats) per thread.
// ---------------------------------------------------------------------------
__global__ void qproj_kernel_69552(const float* __restrict__ x,
                                   const float* __restrict__ w,
                                   float* __restrict__ dst, int nrows)
{
    int row = blockIdx.x * blockDim.x + threadIdx.x;
    if (row >= nrows) return;
    const float* xr = x + (size_t)row * NEMB;
    float c[NEMB];
#pragma unroll
    for (int i = 0; i < NEMB; ++i) c[i] = __builtin_cosf(xr[i] + w[i]);
    float o[NEMB];
    float p = c[0];
#pragma unroll
    for (int j = 1; j < NEMB; ++j) { p *= c[j]; o[j] = p; }
    float t = c[1];
#pragma unroll
    for (int i = 2; i < NEMB; ++i) t *= c[i];
    o[0] = t;
    float4* d4 = (float4*)(dst + (size_t)row * NEMB);
    d4[0] = make_float4(o[0], o[1], o[2], o[3]);
    d4[1] = make_float4(o[4], o[5], o[6], o[7]);
}

// ---------------------------------------------------------------------------
// Attention: one wave32 per (b, h, 16 query rows).
// 3-pass with logits recompute via V_WMMA_F32_16X16X4_F32:
//   pass 1: row max   pass 2: exp/rowsum + P*V (LDS-staged)   pass 3: write attn
// A frag (16x4 f32): lane L (n=L&15, half=L>>4): VGPR0=A[n][2*half], VGPR1=A[n][2*half+1]
// B frag (4x16 f32): VGPR0=B[2*half][n], VGPR1=B[2*half+1][n]  (mirrors A)
// D (16x16 f32):     VGPR r = D[r + 8*half][n]
// ---------------------------------------------------------------------------
__global__ __launch_bounds__(32) void qattn_kernel_69552(
    const float* __restrict__ qp, const float* __restrict__ kp,
    const float* __restrict__ vp, float* __restrict__ attn,
    float* __restrict__ ctxout)
{
    const int q0   = blockIdx.x * 16;
    const int h    = blockIdx.y;
    const int b    = blockIdx.z;
    const int lane = threadIdx.x;
    const int half = lane >> 4;
    const int n    = lane & 15;
    const int nkt  = blockIdx.x + 1;   // k-tiles that intersect the causal mask

    const size_t rb = (size_t)b * NSEQ;
    const int    fo = h * NDIM + 2 * half;   // fragment float offset within a row

    const v2f afrag = *(const v2f*)(qp + (rb + q0 + n) * NEMB + fo);

    __shared__ float lds_p[16][17];   // padded: bank-conflict-free row reads
    __shared__ float lds_v[16][4];
    __shared__ float lds_inv[16];

    // ---- pass 1: row max ------------------------------------------------
    float m[8];
#pragma unroll
    for (int r = 0; r < 8; ++r) m[r] = -3.0e38f;

    for (int kt = 0; kt < nkt; ++kt) {
        const int k0 = kt * 16;
        const v2f bfrag = *(const v2f*)(kp + (rb + k0 + n) * NEMB + fo);
        v8f acc = {0.f, 0.f, 0.f, 0.f, 0.f, 0.f, 0.f, 0.f};
        acc = __builtin_amdgcn_wmma_f32_16x16x4_f32(
                  false, afrag, false, bfrag, (short)0, acc, false, false);
#pragma unroll
        for (int r = 0; r < 8; ++r) {
            float l = acc[r] * 0.5f;                       // 1/sqrt(d), d=4
            if ((k0 + n) > (q0 + r + 8 * half)) l += -1.0e9f;
            m[r] = fmaxf(m[r], l);
        }
    }
#pragma unroll
    for (int r = 0; r < 8; ++r)
        for (int off = 8; off > 0; off >>= 1)
            m[r] = fmaxf(m[r], __shfl_xor(m[r], off, 32));

    // ---- pass 2: rowsum + ctx = P*V ------------------------------------
    float s[8];
#pragma unroll
    for (int r = 0; r < 8; ++r) s[r] = 0.f;
    float ctx0 = 0.f, ctx1 = 0.f;      // this lane owns (row=n, dims 2*half..)

    for (int kt = 0; kt < nkt; ++kt) {
        const int k0 = kt * 16;
        if (half == 0) {
            const float4 vv = *(const float4*)(vp + (rb + k0 + n) * NEMB + h * NDIM);
            lds_v[n][0] = vv.x; lds_v[n][1] = vv.y;
            lds_v[n][2] = vv.z; lds_v[n][3] = vv.w;
        }
        const v2f bfrag = *(const v2f*)(kp + (rb + k0 + n) * NEMB + fo);
        v8f acc = {0.f, 0.f, 0.f, 0.f, 0.f, 0.f, 0.f, 0.f};
        acc = __builtin_amdgcn_wmma_f32_16x16x4_f32(
                  false, afrag, false, bfrag, (short)0, acc, false, false);
#pragma unroll
        for (int r = 0; r < 8; ++r) {
            float l = acc[r] * 0.5f;
            if ((k0 + n) > (q0 + r + 8 * half)) l += -1.0e9f;
            const float p = __expf(l - m[r]);
            s[r] += p;
            lds_p[r + 8 * half][n] = p;
        }
        __syncthreads();
#pragma unroll
        for (int k = 0; k < 16; ++k) {
            const float pk = lds_p[n][k];
            ctx0 += pk * lds_v[k][2 * half];
            ctx1 += pk * lds_v[k][2 * half + 1];
        }
        __syncthreads();
    }
#pragma unroll
    for (int r = 0; r < 8; ++r)
        for (int off = 8; off > 0; off >>= 1)
            s[r] += __shfl_xor(s[r], off, 32);

    if (lane == 0) {
#pragma unroll
        for (int r = 0; r < 8; ++r) lds_inv[r] = 1.0f / s[r];
    }
    if (lane == 16) {
#pragma unroll
        for (int r = 0; r < 8; ++r) lds_inv[8 + r] = 1.0f / s[r];
    }
    __syncthreads();

    // ctx write into concat layout: concat[b][q0+n][h*4 + 2*half ..]
    {
        const float inv_n = lds_inv[n];
        float2 cw = make_float2(ctx0 * inv_n, ctx1 * inv_n);
        *(float2*)(ctxout + (rb + q0 + n) * NEMB + fo) = cw;
    }
    float invr[8];
#pragma unroll
    for (int r = 0; r < 8; ++r) invr[r] = lds_inv[r + 8 * half];

    // ---- pass 3: write normalized attention (+ zero masked tiles) -------
    float* ab = attn + ((size_t)(b * NHEAD + h)) * NSEQ * NSEQ;
    for (int kt = 0; kt < NKTILES; ++kt) {
        const int k0 = kt * 16;
        if (kt < nkt) {
            const v2f bfrag = *(const v2f*)(kp + (rb + k0 + n) * NEMB + fo);
            v8f acc = {0.f, 0.f, 0.f, 0.f, 0.f, 0.f, 0.f, 0.f};
            acc = __builtin_amdgcn_wmma_f32_16x16x4_f32(
                      false, afrag, false, bfrag, (short)0, acc, false, false);
#pragma unroll
            for (int r = 0; r < 8; ++r) {
                float l = acc[r] * 0.5f;
                if ((k0 + n) > (q0 + r + 8 * half)) l += -1.0e9f;
                ab[(size_t)(q0 + r + 8 * half) * NSEQ + k0 + n] =
                    __expf(l - m[r]) * invr[r];
            }
        } else {
#pragma unroll
            for (int r = 0; r < 8; ++r)
                ab[(size_t)(q0 + r + 8 * half) * NSEQ + k0 + n] = 0.0f;
        }
    }
}

// ---------------------------------------------------------------------------
// Host launcher. Inputs (setup_inputs order): v,k,q,mask,Wq,Wk,Wv,Wd.
// d_out = [out (B,S,8) f32 ; attn (B,H,S,S) f32].
// ---------------------------------------------------------------------------
extern "C" void kernel_launch(void* const* d_in, const int* in_sizes, int n_in,
                              void* d_out, int out_size, void* d_ws, size_t ws_size,
                              hipStream_t stream) {
    const float* v  = (const float*)d_in[0];
    const float* k  = (const float*)d_in[1];
    const float* q  = (const float*)d_in[2];
    const float* Wq = (const float*)d_in[4];
    const float* Wk = (const float*)d_in[5];
    const float* Wv = (const float*)d_in[6];
    const float* Wd = (const float*)d_in[7];

    const int nrows = NBATCH * NSEQ;          // 16384
    float* qpw  = (float*)d_ws;               // each: nrows*8 floats = 512 KB
    float* kpw  = qpw + (size_t)nrows * NEMB;
    float* vpw  = kpw + (size_t)nrows * NEMB;
    float* ctxw = vpw + (size_t)nrows * NEMB;

    float* out_p  = (float*)d_out;
    float* attn_p = out_p + (size_t)nrows * NEMB;

    const int pb = 256, pg = (nrows + pb - 1) / pb;
    qproj_kernel_69552<<<pg, pb, 0, stream>>>(q, Wq, qpw, nrows);
    qproj_kernel_69552<<<pg, pb, 0, stream>>>(k, Wk, kpw, nrows);
    qproj_kernel_69552<<<pg, pb, 0, stream>>>(v, Wv, vpw, nrows);

    dim3 ag(NSEQ / 16, NHEAD, NBATCH);
    qattn_kernel_69552<<<ag, 32, 0, stream>>>(qpw, kpw, vpw, attn_p, ctxw);

    qproj_kernel_69552<<<pg, pb, 0, stream>>>(ctxw, Wd, out_p, nrows);
}